// GRU_3100966388317
// MI455X (gfx1250) — compile-verified
//
#include <hip/hip_runtime.h>
#include <hip/hip_bf16.h>
#include <math.h>

// ---------------------------------------------------------------------------
// IBP-GRU for MI455X (gfx1250), fp32 WMMA path (V_WMMA_F32_16X16X4_F32).
//
// Phase 1 (parallel):  px{v,l,u} = ibp_linear(x, Wi, bi) for all (b,t) rows.
//                      3 GEMMs (65536 x 384 x 128) -> d_ws (302 MB).
// Phase 2 (recurrent): 8 persistent WGs, each owns 16 batch rows; Wh in LDS;
//                      per step 3 GEMMs (16 x 384 x 128) via wmma + fused
//                      interval gate math; writes (3,B,T,H) to d_out.
// ---------------------------------------------------------------------------

#define B_  128
#define T_  512
#define D_  128
#define H_  128
#define H3  384
#define NPX ((size_t)B_ * T_ * H3)      // elements per px variant (25165824)
#define OUTV ((size_t)B_ * T_ * H_)     // elements per output variant

typedef float v2f __attribute__((ext_vector_type(2)));
typedef float v4f __attribute__((ext_vector_type(4)));
typedef float v8f __attribute__((ext_vector_type(8)));

static __device__ __forceinline__ v8f wmma_f32x4(v2f a, v2f b, v8f c) {
  // D = A(16x4,f32) * B(4x16,f32) + C(16x16,f32); wave32, EXEC must be all 1s.
  return __builtin_amdgcn_wmma_f32_16x16x4_f32(
      /*neg_a=*/false, a, /*neg_b=*/false, b,
      /*c_mod=*/(short)0, c, /*reuse_a=*/false, /*reuse_b=*/false);
}

static __device__ __forceinline__ float sigmoidf_(float x) {
  return 1.0f / (1.0f + __expf(-x));
}
static __device__ __forceinline__ float min4f(float a, float b, float c, float d) {
  return fminf(fminf(a, b), fminf(c, d));
}
static __device__ __forceinline__ float max4f(float a, float b, float c, float d) {
  return fmaxf(fmaxf(a, b), fmaxf(c, d));
}

// ---------------------------------------------------------------------------
// Phase 1: input projection with IBP center/radius propagation.
// Grid: 4096 blocks (one 16-row tile of the 65536 flattened (b,t) rows each),
// 256 threads = 8 waves; each wave handles 3 column tiles (8*3*16 = 384 cols).
// ---------------------------------------------------------------------------
__global__ __launch_bounds__(256) void ibp_gru_phase1(
    const float* __restrict__ xv, const float* __restrict__ xl,
    const float* __restrict__ xu, const float* __restrict__ Wi,
    const float* __restrict__ bi, float* __restrict__ px) {
  const int lane = threadIdx.x & 31;
  const int wave = threadIdx.x >> 5;
  const int l16  = lane & 15;
  const int half = lane >> 4;

  const int rowBase = blockIdx.x * 16;          // 4096 * 16 = 65536 rows
  const int row     = rowBase + l16;

  const float* avp = xv + (size_t)row * D_ + 2 * half;
  const float* alp = xl + (size_t)row * D_ + 2 * half;
  const float* aup = xu + (size_t)row * D_ + 2 * half;

  const float* bp0 = Wi + (size_t)((wave * 3 + 0) * 16 + l16) * D_ + 2 * half;
  const float* bp1 = Wi + (size_t)((wave * 3 + 1) * 16 + l16) * D_ + 2 * half;
  const float* bp2 = Wi + (size_t)((wave * 3 + 2) * 16 + l16) * D_ + 2 * half;

  v8f accv0 = {}, accv1 = {}, accv2 = {};
  v8f accm0 = {}, accm1 = {}, accm2 = {};
  v8f accr0 = {}, accr1 = {}, accr2 = {};

#pragma unroll 4
  for (int c = 0; c < 32; ++c) {                // K = 128 in chunks of 4
    const int k = 4 * c;
    v2f av = *(const v2f*)(avp + k);
    v2f lo = *(const v2f*)(alp + k);
    v2f hi = *(const v2f*)(aup + k);
    v2f am = (hi + lo) * 0.5f;                  // center
    v2f ar = (hi - lo) * 0.5f;                  // radius

    v2f b0 = *(const v2f*)(bp0 + k);
    v2f b1 = *(const v2f*)(bp1 + k);
    v2f b2 = *(const v2f*)(bp2 + k);
    v2f a0 = {fabsf(b0.x), fabsf(b0.y)};
    v2f a1 = {fabsf(b1.x), fabsf(b1.y)};
    v2f a2 = {fabsf(b2.x), fabsf(b2.y)};

    accv0 = wmma_f32x4(av, b0, accv0);
    accv1 = wmma_f32x4(av, b1, accv1);
    accv2 = wmma_f32x4(av, b2, accv2);
    accm0 = wmma_f32x4(am, b0, accm0);
    accm1 = wmma_f32x4(am, b1, accm1);
    accm2 = wmma_f32x4(am, b2, accm2);
    accr0 = wmma_f32x4(ar, a0, accr0);
    accr1 = wmma_f32x4(ar, a1, accr1);
    accr2 = wmma_f32x4(ar, a2, accr2);
  }

  const int r0 = rowBase + 8 * half;            // C layout: VGPR i -> row i+8*half
#pragma unroll
  for (int j = 0; j < 3; ++j) {
    const int col = (wave * 3 + j) * 16 + l16;
    const float bias = bi[col];
    const v8f* pv = (j == 0) ? &accv0 : (j == 1) ? &accv1 : &accv2;
    const v8f* pm = (j == 0) ? &accm0 : (j == 1) ? &accm1 : &accm2;
    const v8f* pr = (j == 0) ? &accr0 : (j == 1) ? &accr1 : &accr2;
#pragma unroll
    for (int i = 0; i < 8; ++i) {
      const size_t off = (size_t)(r0 + i) * H3 + col;
      const float m = (*pm)[i] + bias;
      const float r = (*pr)[i];
      px[off]           = (*pv)[i] + bias;      // pv
      px[NPX + off]     = m - r;                // pl
      px[2 * NPX + off] = m + r;                // pu
    }
  }
}

// ---------------------------------------------------------------------------
// Phase 2: persistent recurrent kernel. 8 blocks x 256 threads (8 waves).
// LDS (dynamic, 302592 B total, pitched to kill bank conflicts):
//   sWh [384][132]   Wh row-major, pitch 132 (132 % 64 == 4 -> conflict-free)
//   sH  [3][16][132] hidden state as (val, mu, rad)
//   sQ  [3][16][388] recurrent preactivations q = h @ Wh^T (raw, bias later)
// ---------------------------------------------------------------------------
#define WH_PITCH 132
#define SH_PITCH 132
#define SQ_PITCH 388
#define SWH_F (384 * WH_PITCH)          // 50688 floats
#define SH_F  (3 * 16 * SH_PITCH)       //  6336 floats
#define SQ_F  (3 * 16 * SQ_PITCH)       // 18624 floats
#define SMEM_BYTES ((SWH_F + SH_F + SQ_F) * 4)   // 302592 B <= 320 KB WGP LDS

__global__ __launch_bounds__(256) void ibp_gru_phase2(
    const float* __restrict__ px, const float* __restrict__ h0,
    const float* __restrict__ Wh, const float* __restrict__ bh,
    float* __restrict__ out) {
  extern __shared__ float smem[];
  float* sWh = smem;
  float* sH  = smem + SWH_F;
  float* sQ  = smem + SWH_F + SH_F;

  const int tid  = threadIdx.x;
  const int lane = tid & 31;
  const int wave = tid >> 5;
  const int l16  = lane & 15;
  const int half = lane >> 4;
  const int bBase = blockIdx.x * 16;            // 8 blocks * 16 = 128 batch rows

  // Preload Wh (384 x 128) into padded LDS, 128-bit stores.
  for (int idx = tid; idx < 384 * 32; idx += 256) {
    const int n  = idx >> 5;
    const int kg = (idx & 31) << 2;
    *(v4f*)(sWh + n * WH_PITCH + kg) = *(const v4f*)(Wh + (size_t)n * H_ + kg);
  }
  // Init hidden triple: hv = hmu = h0, hr = 0.
  for (int e = tid; e < 16 * H_; e += 256) {
    const int r = e >> 7, j = e & 127;
    const float v = h0[(size_t)(bBase + r) * H_ + j];
    sH[0 * 16 * SH_PITCH + r * SH_PITCH + j] = v;
    sH[1 * 16 * SH_PITCH + r * SH_PITCH + j] = v;
    sH[2 * 16 * SH_PITCH + r * SH_PITCH + j] = 0.0f;
  }
  __syncthreads();

  for (int t = 0; t < T_; ++t) {
    // ---- GEMM: q[g](16x384) = hA[g](16x128) @ Wh^T ; g: 0=val 1=mu 2=rad ---
    for (int tile = wave; tile < 72; tile += 8) {     // 9 tiles per wave
      const int g  = tile / 24;
      const int nt = tile % 24;
      const float* aB = sH + g * 16 * SH_PITCH + l16 * SH_PITCH + 2 * half;
      const float* bB = sWh + (nt * 16 + l16) * WH_PITCH + 2 * half;
      v8f acc = {};
#pragma unroll 8
      for (int c = 0; c < 32; ++c) {
        v2f a = *(const v2f*)(aB + 4 * c);
        v2f b = *(const v2f*)(bB + 4 * c);
        if (g == 2) { b.x = fabsf(b.x); b.y = fabsf(b.y); }   // |Wh| for radius
        acc = wmma_f32x4(a, b, acc);
      }
      float* qp = sQ + g * 16 * SQ_PITCH + (8 * half) * SQ_PITCH + nt * 16 + l16;
#pragma unroll
      for (int i = 0; i < 8; ++i) qp[i * SQ_PITCH] = acc[i];
    }
    __syncthreads();

    // ---- fused interval gate math (16 rows x 128 cols, 8 elems/thread) ----
    for (int e = tid; e < 16 * H_; e += 256) {
      const int r = e >> 7, j = e & 127;
      const size_t pOff = ((size_t)(bBase + r) * T_ + t) * H3;
      const int qRow = r * SQ_PITCH;

      // q triples (add bh; interval: l = mu - rad, u = mu + rad)
      const float bhr = bh[j], bhz = bh[H_ + j], bhn = bh[2 * H_ + j];
      float qv, qm, qr;
      qv = sQ[qRow + j] + bhr;
      qm = sQ[16 * SQ_PITCH + qRow + j] + bhr;
      qr = sQ[32 * SQ_PITCH + qRow + j];
      const float qvr = qv, qlr = qm - qr, qur = qm + qr;
      qv = sQ[qRow + H_ + j] + bhz;
      qm = sQ[16 * SQ_PITCH + qRow + H_ + j] + bhz;
      qr = sQ[32 * SQ_PITCH + qRow + H_ + j];
      const float qvz = qv, qlz = qm - qr, quz = qm + qr;
      qv = sQ[qRow + 2 * H_ + j] + bhn;
      qm = sQ[16 * SQ_PITCH + qRow + 2 * H_ + j] + bhn;
      qr = sQ[32 * SQ_PITCH + qRow + 2 * H_ + j];
      const float qvn = qv, qln = qm - qr, qun = qm + qr;

      // p triples from phase 1 (bias already folded)
      const float pvr = px[pOff + j];
      const float plr = px[NPX + pOff + j];
      const float pur = px[2 * NPX + pOff + j];
      const float pvz = px[pOff + H_ + j];
      const float plz = px[NPX + pOff + H_ + j];
      const float puz = px[2 * NPX + pOff + H_ + j];
      const float pvn = px[pOff + 2 * H_ + j];
      const float pln = px[NPX + pOff + 2 * H_ + j];
      const float pun = px[2 * NPX + pOff + 2 * H_ + j];

      if (t + 1 < T_) {                 // pull next step's p rows toward L2/L0
        __builtin_prefetch(px + pOff + H3 + j, 0, 0);
        __builtin_prefetch(px + NPX + pOff + H3 + j, 0, 0);
        __builtin_prefetch(px + 2 * NPX + pOff + H3 + j, 0, 0);
      }

      // gates (sigmoid monotone -> endpoints)
      const float rv = sigmoidf_(pvr + qvr);
      const float rl = sigmoidf_(plr + qlr);
      const float ru = sigmoidf_(pur + qur);
      const float zv = sigmoidf_(pvz + qvz);
      const float zl = sigmoidf_(plz + qlz);
      const float zu = sigmoidf_(puz + quz);

      // candidate: c = r * q_n (4-corner), n = tanh(p_n + c)
      const float cv  = rv * qvn;
      const float clo = min4f(rl * qln, rl * qun, ru * qln, ru * qun);
      const float chi = max4f(rl * qln, rl * qun, ru * qln, ru * qun);
      const float nv = tanhf(pvn + cv);
      const float nl = tanhf(pln + clo);
      const float nu = tanhf(pun + chi);

      // old hidden interval from (val, mu, rad)
      const float hvv = sH[0 * 16 * SH_PITCH + r * SH_PITCH + j];
      const float hm  = sH[1 * 16 * SH_PITCH + r * SH_PITCH + j];
      const float hr_ = sH[2 * 16 * SH_PITCH + r * SH_PITCH + j];
      const float hl = hm - hr_, hu = hm + hr_;

      // a = z * h (4-corner)
      const float av  = zv * hvv;
      const float alo = min4f(zl * hl, zl * hu, zu * hl, zu * hu);
      const float ahi = max4f(zl * hl, zl * hu, zu * hl, zu * hu);
      // b = (1-z) * n ; interval of (1-z) flips bounds: (1-zu, 1-zl)
      const float wv = 1.0f - zv, wlo = 1.0f - zu, whi = 1.0f - zl;
      const float bv  = wv * nv;
      const float blo = min4f(wlo * nl, wlo * nu, whi * nl, whi * nu);
      const float bhi = max4f(wlo * nl, wlo * nu, whi * nl, whi * nu);

      const float hnv = av + bv;
      const float hnl = alo + blo;
      const float hnu = ahi + bhi;

      // write output (3, B, T, H)
      const size_t oOff = ((size_t)(bBase + r) * T_ + t) * H_ + j;
      out[oOff]            = hnv;
      out[OUTV + oOff]     = hnl;
      out[2 * OUTV + oOff] = hnu;

      // in-place state update (same (r,j) owned by this thread only)
      sH[0 * 16 * SH_PITCH + r * SH_PITCH + j] = hnv;
      sH[1 * 16 * SH_PITCH + r * SH_PITCH + j] = 0.5f * (hnl + hnu);
      sH[2 * 16 * SH_PITCH + r * SH_PITCH + j] = 0.5f * (hnu - hnl);
    }
    __syncthreads();
  }
}

// ---------------------------------------------------------------------------
extern "C" void kernel_launch(void* const* d_in, const int* in_sizes, int n_in,
                              void* d_out, int out_size, void* d_ws, size_t ws_size,
                              hipStream_t stream) {
  (void)in_sizes; (void)n_in; (void)out_size; (void)ws_size;
  const float* xv = (const float*)d_in[0];
  const float* xl = (const float*)d_in[1];
  const float* xu = (const float*)d_in[2];
  const float* h0 = (const float*)d_in[3];
  const float* Wi = (const float*)d_in[4];
  const float* bi = (const float*)d_in[5];
  const float* Wh = (const float*)d_in[6];
  const float* bh = (const float*)d_in[7];
  float* out = (float*)d_out;
  float* px  = (float*)d_ws;   // requires 3 * 65536 * 384 * 4 B = 302 MB

  ibp_gru_phase1<<<dim3(4096), dim3(256), 0, stream>>>(xv, xl, xu, Wi, bi, px);
  ibp_gru_phase2<<<dim3(8), dim3(256), SMEM_BYTES, stream>>>(px, h0, Wh, bh, out);
}